// DenoisingSharpening_56298431316325
// MI455X (gfx1250) — compile-verified
//
#include <hip/hip_runtime.h>
#include <cstdint>
#include <cstddef>

// ---------------------------------------------------------------------------
// 5x5 bilateral filter, fp32, NHWC (16,512,512,3).
// VALU/trans-bound op (≈300 VALU + 25 v_exp_f32 per pixel vs ~24 B/pixel of
// HBM traffic), so the plan is: stage the halo tile in LDS with gfx1250 async
// global->LDS copies, then do the 25-tap loop out of LDS with one FMA + one
// v_exp_f32 per tap (spatial weight folded into the exp2 argument).
// ---------------------------------------------------------------------------

#define TX   32               // tile width  (one wave32 per row)
#define TY   8                // tile height (256 threads = 8 waves)
#define HALO 2
#define RX   (TX + 2 * HALO)  // 36 region cols
#define RY   (TY + 2 * HALO)  // 12 region rows
#define NPIX (RX * RY)        // 432 region pixels

__device__ __forceinline__ int reflect_idx(int i, int n) {
    // jnp.pad mode='reflect': -1 -> 1, -2 -> 2, n -> n-2, n+1 -> n-3
    if (i < 0)  i = -i;
    if (i >= n) i = 2 * n - 2 - i;
    return i;
}

__global__ __launch_bounds__(TX * TY)
void bilateral5x5_kernel(const float* __restrict__ img,
                         float* __restrict__ out,
                         int H, int W) {
    // 3 floats (12 B) per region pixel, matching the global NHWC layout so the
    // async copy's shared `offset:` (added to BOTH lds and global addresses)
    // lines up on both sides.
    __shared__ float tile[NPIX * 3];

    const int bx = blockIdx.x * TX;
    const int by = blockIdx.y * TY;
    const int b  = blockIdx.z;
    const int lx = threadIdx.x;
    const int ly = threadIdx.y;
    const int t  = ly * TX + lx;

    const float* imgB = img + (size_t)b * H * W * 3;

    // Low 32 bits of a generic pointer to __shared__ == LDS byte offset.
    const unsigned ldsBase = (unsigned)(uintptr_t)(&tile[0]);

    // ---- stage (TX+4)x(TY+4) halo region into LDS with async copies -------
    for (int p = t; p < NPIX; p += TX * TY) {
        const int ry = p / RX;
        const int rx = p - ry * RX;
        const int gy = reflect_idx(by - HALO + ry, H);
        const int gx = reflect_idx(bx - HALO + rx, W);
        const float* gp = imgB + ((size_t)gy * W + gx) * 3;
        const unsigned la = ldsBase + (unsigned)p * 12u;
        // GLOBAL_LOAD_ASYNC_TO_LDS_B32: vdst = per-lane LDS byte address,
        // vaddr = 64-bit global address, tracked by ASYNCcnt. Three b32 ops
        // keep both LDS and global sides 4-byte aligned (12 B/pixel stride).
        asm volatile("global_load_async_to_lds_b32 %0, %1, off"
                     :: "v"(la), "v"(gp) : "memory");
        asm volatile("global_load_async_to_lds_b32 %0, %1, off offset:4"
                     :: "v"(la), "v"(gp) : "memory");
        asm volatile("global_load_async_to_lds_b32 %0, %1, off offset:8"
                     :: "v"(la), "v"(gp) : "memory");
    }
    asm volatile("s_wait_asynccnt 0x0" ::: "memory");
    __syncthreads();

    // ---- 25-tap bilateral out of LDS ---------------------------------------
    // log2 of the normalized 1-D spatial Gaussian (sigma_s = 1, k = 5):
    //   g = [e^-2, e^-0.5, 1, e^-0.5, e^-2] / 2.4837318859
    const float lw[5] = {-4.1978994f, -2.0338569f, -1.3125093f,
                         -2.0338569f, -4.1978994f};
    // K = -(0.5 / sigma_r^2) * log2(e) = -312.5 * 1.4426950409
    const float K = -450.8422003f;

    const int cbase = ((ly + HALO) * RX + (lx + HALO)) * 3;
    const float c0 = tile[cbase + 0];
    const float c1 = tile[cbase + 1];
    const float c2 = tile[cbase + 2];

    float n0 = 0.f, n1 = 0.f, n2 = 0.f, den = 0.f;
#pragma unroll
    for (int dy = 0; dy < 5; ++dy) {
#pragma unroll
        for (int dx = 0; dx < 5; ++dx) {
            const int q = ((ly + dy) * RX + (lx + dx)) * 3;
            const float r  = tile[q + 0];
            const float g  = tile[q + 1];
            const float bl = tile[q + 2];
            const float dr = r - c0, dg = g - c1, db = bl - c2;
            const float d2 = dr * dr + dg * dg + db * db;
            // w = sk[dy][dx] * exp(-d2 / (2*sigma_r^2))
            //   = exp2(K*d2 + log2(skY) + log2(skX))   -> 1 FMA + 1 v_exp_f32
            const float w = __builtin_amdgcn_exp2f(fmaf(K, d2, lw[dy] + lw[dx]));
            n0 = fmaf(w, r,  n0);
            n1 = fmaf(w, g,  n1);
            n2 = fmaf(w, bl, n2);
            den += w;
        }
    }

    const float inv = __builtin_amdgcn_rcpf(den);
    const size_t o = (((size_t)b * H + (by + ly)) * W + (bx + lx)) * 3;
    out[o + 0] = fminf(fmaxf(n0 * inv, 0.f), 1.f);
    out[o + 1] = fminf(fmaxf(n1 * inv, 0.f), 1.f);
    out[o + 2] = fminf(fmaxf(n2 * inv, 0.f), 1.f);
}

extern "C" void kernel_launch(void* const* d_in, const int* in_sizes, int n_in,
                              void* d_out, int out_size, void* d_ws, size_t ws_size,
                              hipStream_t stream) {
    (void)n_in; (void)out_size; (void)d_ws; (void)ws_size;
    const float* img = (const float*)d_in[0];
    float*       out = (float*)d_out;

    const int H = 512, W = 512;
    const int B = in_sizes[0] / (H * W * 3);   // 16

    dim3 grid(W / TX, H / TY, B);              // (16, 64, 16)
    dim3 block(TX, TY, 1);                     // 256 threads = 8 wave32
    hipLaunchKernelGGL(bilateral5x5_kernel, grid, block, 0, stream,
                       img, out, H, W);
}